// GCN_61460982005902
// MI455X (gfx1250) — compile-verified
//
#include <hip/hip_runtime.h>
#include <math.h>

#define NEG_SLOPE 0.1f
#define WT_STRIDE 66   // padded LDS stride (floats) for transposed weights: kills bank conflicts

typedef float v2f __attribute__((ext_vector_type(2)));
typedef float v8f __attribute__((ext_vector_type(8)));

__device__ __forceinline__ float lrelu(float v) {
    return v > 0.0f ? v : NEG_SLOPE * v;
}

// ---------------------------------------------------------------------------
// Encoder layer 1: h1[obj][c] = leakyrelu(x[obj] @ W1_type + b1_type)
// One thread per (object, output column). K is only 3 or 6.
// ---------------------------------------------------------------------------
__global__ void enc1_kernel(const float* __restrict__ xg, const float* __restrict__ xl,
                            const float* __restrict__ xo, const float* __restrict__ xe,
                            const float* __restrict__ Wg, const float* __restrict__ bg,
                            const float* __restrict__ Wl, const float* __restrict__ bl,
                            const float* __restrict__ Wo, const float* __restrict__ bo,
                            const float* __restrict__ We, const float* __restrict__ be,
                            float* __restrict__ out,
                            int nGen, int nLoad, int nOr, int nTot) {
    int idx = blockIdx.x * blockDim.x + threadIdx.x;
    if (idx >= nTot * 64) return;
    int obj = idx >> 6;
    int c   = idx & 63;

    const float* x; const float* W; const float* b; int dim;
    if (obj < nGen)                    { x = xg + (size_t)obj * 3;                      W = Wg; b = bg; dim = 3; }
    else if (obj < nGen + nLoad)       { x = xl + (size_t)(obj - nGen) * 3;             W = Wl; b = bl; dim = 3; }
    else if (obj < nGen + nLoad + nOr) { x = xo + (size_t)(obj - nGen - nLoad) * 6;     W = Wo; b = bo; dim = 6; }
    else                               { x = xe + (size_t)(obj - nGen - nLoad - nOr) * 6; W = We; b = be; dim = 6; }

    float s = b[c];
    for (int k = 0; k < dim; ++k) s += x[k] * W[k * 64 + c];
    out[(size_t)obj * 64 + c] = lrelu(s);
}

// ---------------------------------------------------------------------------
// WMMA GEMM: out = leakyrelu(A1 @ W1 [+ A2 @ W2] + bias)
//   A1,A2 : rows x 64 (row-major f32), W1,W2 : 64 x 64, bias : 64
// Block = 256 threads = 8 waves; each wave owns a 16-row slab and 4 column
// tiles of 16x16 computed with V_WMMA_F32_16X16X4_F32 (16 k-steps).
// DUAL is a compile-time flag so the single-matrix variant has no per-step
// branch and no second LDS weight buffer.
// Weights are staged TRANSPOSED in LDS (Wt[c][k], padded stride) so each
// B fragment is one contiguous ds_load_b64 without bank conflicts.
// A rows are clamped (not predicated): an out-of-range A row only affects
// the matching out row, whose store is guarded -> no EXEC juggling.
// In-place (out aliasing A1/A2) is safe: a wave reads/writes only its slab.
// ---------------------------------------------------------------------------
template <bool DUAL>
__global__ void gemm64_kernel(const float* A1, const float* W1g,
                              const float* A2, const float* W2g,
                              const float* bias, float* out, int rows) {
    __shared__ float W1t[64 * WT_STRIDE];
    __shared__ float W2t[DUAL ? 64 * WT_STRIDE : 1];
    __shared__ float bs[64];

    const int tid = threadIdx.x;
    for (int idx = tid; idx < 4096; idx += blockDim.x) {
        int k = idx >> 6, c = idx & 63;          // W row-major: W[k][c]
        W1t[c * WT_STRIDE + k] = W1g[idx];       // store transposed
    }
    if (DUAL) {
        for (int idx = tid; idx < 4096; idx += blockDim.x) {
            int k = idx >> 6, c = idx & 63;
            W2t[c * WT_STRIDE + k] = W2g[idx];
        }
    }
    if (tid < 64) bs[tid] = bias[tid];
    __syncthreads();

    const int wave = tid >> 5;
    const int lane = tid & 31;
    const int half = lane >> 4;      // 0: lanes 0-15, 1: lanes 16-31
    const int ln   = lane & 15;
    const int row0 = (blockIdx.x * (blockDim.x >> 5) + wave) * 16;
    if (row0 >= rows) return;        // wave-uniform exit

    v8f acc[4];
#pragma unroll
    for (int ct = 0; ct < 4; ++ct)
#pragma unroll
        for (int i = 0; i < 8; ++i) acc[ct][i] = 0.0f;

    const int arow = row0 + ln;
    const int crow = (arow < rows) ? arow : (rows - 1);   // clamp: no predication needed
    const float* a1p = A1 + (size_t)crow * 64;
    const float* a2p = DUAL ? (A2 + (size_t)crow * 64) : nullptr;

#pragma unroll
    for (int kk = 0; kk < 64; kk += 4) {
        const int k0 = kk + 2 * half;  // A 16x4 f32: lanes 0-15 hold K=k,k+1; lanes 16-31 hold K=k+2,k+3
        const v2f a = *(const v2f*)(a1p + k0);             // global_load_b64
#pragma unroll
        for (int ct = 0; ct < 4; ++ct) {
            const v2f b = *(const v2f*)(&W1t[(ct * 16 + ln) * WT_STRIDE + k0]);  // ds_load_b64
            acc[ct] = __builtin_amdgcn_wmma_f32_16x16x4_f32(
                false, a, false, b, (short)0, acc[ct], false, false);
        }
        if (DUAL) {
            const v2f a2 = *(const v2f*)(a2p + k0);
#pragma unroll
            for (int ct = 0; ct < 4; ++ct) {
                const v2f b = *(const v2f*)(&W2t[(ct * 16 + ln) * WT_STRIDE + k0]);
                acc[ct] = __builtin_amdgcn_wmma_f32_16x16x4_f32(
                    false, a2, false, b, (short)0, acc[ct], false, false);
            }
        }
    }

    // C/D 16x16 f32 layout: VGPR i holds row i (lanes 0-15) / row i+8 (lanes 16-31)
    if (row0 + 16 <= rows) {         // full slab: unguarded stores (wave-uniform branch)
#pragma unroll
        for (int ct = 0; ct < 4; ++ct) {
            const int c = ct * 16 + ln;
            const float bv = bs[c];
#pragma unroll
            for (int i = 0; i < 8; ++i) {
                const int r = row0 + i + 8 * half;
                out[(size_t)r * 64 + c] = lrelu(acc[ct][i] + bv);
            }
        }
    } else {                         // partial trailing slab
#pragma unroll
        for (int ct = 0; ct < 4; ++ct) {
            const int c = ct * 16 + ln;
            const float bv = bs[c];
#pragma unroll
            for (int i = 0; i < 8; ++i) {
                const int r = row0 + i + 8 * half;
                if (r < rows) out[(size_t)r * 64 + c] = lrelu(acc[ct][i] + bv);
            }
        }
    }
}

// ---------------------------------------------------------------------------
// Gather: h[i] = enc[ptv[i]]   (float4 granularity)
// ---------------------------------------------------------------------------
__global__ void gather_kernel(const float* __restrict__ enc, const int* __restrict__ ptv,
                              float* __restrict__ h, int n) {
    int idx = blockIdx.x * blockDim.x + threadIdx.x;
    if (idx >= n * 16) return;
    int i = idx >> 4, c = idx & 15;
    int s = ptv[i];
    ((float4*)(h + (size_t)i * 64))[c] = ((const float4*)(enc + (size_t)s * 64))[c];
}

// ---------------------------------------------------------------------------
// Zero a float buffer (float4 granularity)
// ---------------------------------------------------------------------------
__global__ void zero_kernel(float4* __restrict__ p, int n4) {
    int i = blockIdx.x * blockDim.x + threadIdx.x;
    if (i < n4) p[i] = make_float4(0.f, 0.f, 0.f, 0.f);
}

// ---------------------------------------------------------------------------
// Scatter-add: agg[dst[e]] += h[src[e]]   (16 threads per edge, float4 each)
// h/agg are 25.6 MB each -> L2-resident (192 MB); atomics resolve at L2.
// ---------------------------------------------------------------------------
__global__ void scatter_kernel(const int* __restrict__ src, const int* __restrict__ dst,
                               const float* __restrict__ h, float* __restrict__ agg,
                               int nE) {
    long long idx = (long long)blockIdx.x * blockDim.x + threadIdx.x;
    if (idx >= (long long)nE * 16) return;
    int e = (int)(idx >> 4);
    int c = (int)(idx & 15);
    int s = src[e], d = dst[e];
    if (c == 0 && e + 1024 < nE)
        __builtin_prefetch(h + (size_t)src[e + 1024] * 64, 0, 0);  // global_prefetch_b8
    const float4 v = ((const float4*)(h + (size_t)s * 64))[c];
    float* p = agg + (size_t)d * 64 + c * 4;
    atomicAdd(p + 0, v.x);
    atomicAdd(p + 1, v.y);
    atomicAdd(p + 2, v.z);
    atomicAdd(p + 3, v.w);
}

// ---------------------------------------------------------------------------
// Final layer: out[i] = sigmoid(agg[i].Wl + h[i].Wr + b)   (H -> 1)
// ---------------------------------------------------------------------------
__global__ void final_kernel(const float* __restrict__ agg, const float* __restrict__ h,
                             const float* __restrict__ Wl, const float* __restrict__ Wr,
                             const float* __restrict__ b, float* __restrict__ out, int n) {
    int i = blockIdx.x * blockDim.x + threadIdx.x;
    if (i >= n) return;
    const float* ap = agg + (size_t)i * 64;
    const float* hp = h + (size_t)i * 64;
    float s = b[0];
#pragma unroll 8
    for (int k = 0; k < 64; ++k) s += ap[k] * Wl[k] + hp[k] * Wr[k];
    out[i] = 1.0f / (1.0f + __expf(-s));
}

// ---------------------------------------------------------------------------
extern "C" void kernel_launch(void* const* d_in, const int* in_sizes, int n_in,
                              void* d_out, int out_size, void* d_ws, size_t ws_size,
                              hipStream_t stream) {
    const float* xg  = (const float*)d_in[0];
    const float* xl  = (const float*)d_in[1];
    const float* xo  = (const float*)d_in[2];
    const float* xe  = (const float*)d_in[3];
    const int*   ei  = (const int*)d_in[4];
    const int*   ptv = (const int*)d_in[5];
    const float *Wg1 = (const float*)d_in[6],  *bg1 = (const float*)d_in[7];
    const float *Wg2 = (const float*)d_in[8],  *bg2 = (const float*)d_in[9];
    const float *Wl1 = (const float*)d_in[10], *bl1 = (const float*)d_in[11];
    const float *Wl2 = (const float*)d_in[12], *bl2 = (const float*)d_in[13];
    const float *Wo1 = (const float*)d_in[14], *bo1 = (const float*)d_in[15];
    const float *Wo2 = (const float*)d_in[16], *bo2 = (const float*)d_in[17];
    const float *We1 = (const float*)d_in[18], *be1 = (const float*)d_in[19];
    const float *We2 = (const float*)d_in[20], *be2 = (const float*)d_in[21];
    const float *WlH = (const float*)d_in[22];
    const float *WrH = (const float*)d_in[23];
    const float *bH  = (const float*)d_in[24];
    const float *WlL = (const float*)d_in[25];
    const float *WrL = (const float*)d_in[26];
    const float *bL  = (const float*)d_in[27];

    const int nGen  = in_sizes[0] / 3;
    const int nLoad = in_sizes[1] / 3;
    const int nOr   = in_sizes[2] / 6;
    const int nEx   = in_sizes[3] / 6;
    const int nE    = in_sizes[4] / 2;
    const int Ntot  = in_sizes[5];
    const int* srcI = ei;         // edge_index[0]
    const int* dstI = ei + nE;    // edge_index[1]

    // Workspace: two N x 64 f32 maps (L2-resident; 2 x 25.6 MB)
    float* enc = (float*)d_ws;                      // encoder buffer, later reused as agg
    float* h   = enc + (size_t)Ntot * 64;           // live features
    float* agg = enc;

    // 1) encoder layer 1 (K=3/6)
    {
        int tot = Ntot * 64;
        enc1_kernel<<<(tot + 255) / 256, 256, 0, stream>>>(
            xg, xl, xo, xe, Wg1, bg1, Wl1, bl1, Wo1, bo1, We1, be1,
            enc, nGen, nLoad, nOr, Ntot);
    }
    // 2) encoder layer 2 per type: WMMA GEMM 64x64, in-place (single-matrix variant)
    {
        const float* W2s[4] = {Wg2, Wl2, Wo2, We2};
        const float* b2s[4] = {bg2, bl2, bo2, be2};
        const int    cnt[4] = {nGen, nLoad, nOr, nEx};
        long off = 0;
        for (int t = 0; t < 4; ++t) {
            float* p = enc + off * 64;
            gemm64_kernel<false><<<(cnt[t] + 127) / 128, 256, 0, stream>>>(
                p, W2s[t], nullptr, nullptr, b2s[t], p, cnt[t]);
            off += cnt[t];
        }
    }
    // 3) gather via object_ptv
    gather_kernel<<<(Ntot * 16 + 255) / 256, 256, 0, stream>>>(enc, ptv, h, Ntot);

    // 4) seven hidden SAGE layers: zero agg -> scatter-add -> dual WMMA GEMM (in-place on h)
    const long long scTot = (long long)nE * 16;
    const int scGrid = (int)((scTot + 255) / 256);
    for (int l = 0; l < 7; ++l) {
        zero_kernel<<<(Ntot * 16 + 255) / 256, 256, 0, stream>>>((float4*)agg, Ntot * 16);
        scatter_kernel<<<scGrid, 256, 0, stream>>>(srcI, dstI, h, agg, nE);
        gemm64_kernel<true><<<(Ntot + 127) / 128, 256, 0, stream>>>(
            agg, WlH + (size_t)l * 4096, h, WrH + (size_t)l * 4096,
            bH + (size_t)l * 64, h, Ntot);
    }

    // 5) output layer: zero -> scatter -> (H->1) dot + sigmoid
    zero_kernel<<<(Ntot * 16 + 255) / 256, 256, 0, stream>>>((float4*)agg, Ntot * 16);
    scatter_kernel<<<scGrid, 256, 0, stream>>>(srcI, dstI, h, agg, nE);
    final_kernel<<<(Ntot + 255) / 256, 256, 0, stream>>>(agg, h, WlL, WrL, bL,
                                                         (float*)d_out, Ntot);
}